// HMM_10385230922019
// MI455X (gfx1250) — compile-verified
//
#include <hip/hip_runtime.h>
#include <hip/hip_bf16.h>
#include <stdint.h>

#ifndef __has_builtin
#define __has_builtin(x) 0
#endif

#if __has_builtin(__builtin_amdgcn_global_load_async_to_lds_b32)
#define HMM_ASYNC_LDS 1
#else
#define HMM_ASYNC_LDS 0
#endif

#if __has_builtin(__builtin_amdgcn_mov_dpp)
#define HMM_DPP 1
#else
#define HMM_DPP 0
#endif

#if __has_builtin(__builtin_amdgcn_exp2f)
#define HMM_EXP2(x) __builtin_amdgcn_exp2f(x)   // raw v_exp_f32 (base-2)
#else
#define HMM_EXP2(x) exp2f(x)
#endif

#if __has_builtin(__builtin_amdgcn_logf)
#define HMM_LOG2(x) __builtin_amdgcn_logf(x)    // raw v_log_f32 (base-2)
#else
#define HMM_LOG2(x) __log2f(x)
#endif

#define HMM_S 4
#define HMM_T 512
#define HMM_V 6000

typedef __attribute__((address_space(1))) int* hmm_gptr;
typedef __attribute__((address_space(3))) int* hmm_lptr;

__device__ __forceinline__ void hmm_wait_async_lds() {
#if HMM_ASYNC_LDS
#if __has_builtin(__builtin_amdgcn_s_wait_asynccnt)
  __builtin_amdgcn_s_wait_asynccnt(0);
#else
  asm volatile("s_wait_asynccnt 0" ::: "memory");
#endif
#endif
}

// Intra-quad xor-gather: lane j receives the value held by lane j^K.
template <int K>
__device__ __forceinline__ float hmm_qxor(float x) {
#if HMM_DPP
  // quad_perm DPP controls for xor-1/2/3 within each group of 4 lanes.
  constexpr int CTRL = (K == 1) ? 0xB1    // {1,0,3,2}
                     : (K == 2) ? 0x4E    // {2,3,0,1}
                                : 0x1B;   // {3,2,1,0}
  int r = __builtin_amdgcn_mov_dpp(__builtin_bit_cast(int, x), CTRL, 0xF, 0xF, true);
  return __builtin_bit_cast(float, r);
#else
  return __shfl_xor(x, K, 32);
#endif
}

// One HMM forward step for state j (this lane); alpha values for the other
// states of this sequence live in the other 3 lanes of the quad.
// Weights are pre-permuted per lane: Ws = log_A[j][j], Wxk = log_A[j^k][j].
__device__ __forceinline__ float hmm_step(float a,
                                          float Ws, float Wx1, float Wx2, float Wx3,
                                          float e, float m) {
  float b1 = hmm_qxor<1>(a);
  float b2 = hmm_qxor<2>(a);
  float b3 = hmm_qxor<3>(a);
  float x0 = a + Ws, x1 = b1 + Wx1, x2 = b2 + Wx2, x3 = b3 + Wx3;
  float mx = fmaxf(fmaxf(x0, x1), fmaxf(x2, x3));
  const float L2E = 1.4426950408889634f;   // log2(e)
  const float LN2 = 0.6931471805599453f;   // ln(2)
  float mm = mx * L2E;
  // exp((x-mx)) computed as exp2(x*log2e - mx*log2e); args <= ~0, sum >= ~1.
  float s = HMM_EXP2(fmaf(x0, L2E, -mm)) + HMM_EXP2(fmaf(x1, L2E, -mm)) +
            HMM_EXP2(fmaf(x2, L2E, -mm)) + HMM_EXP2(fmaf(x3, L2E, -mm));
  float nv = fmaf(HMM_LOG2(s), LN2, mx) + e;   // logsumexp + emission
  return m * nv + (1.0f - m) * a;              // exact when m == 1 (matches reference)
}

__global__ __launch_bounds__(256) void hmm_fwd_kernel(
    const int* __restrict__ Y, const float* __restrict__ mask,
    const float* __restrict__ log_pi, const float* __restrict__ log_A,
    const float* __restrict__ log_B, float* __restrict__ out) {
  // Emission table transposed to [vocab][state] in LDS: 96000 bytes.
  __shared__ float ldsB[HMM_S * HMM_V];

  for (int idx = threadIdx.x; idx < HMM_S * HMM_V; idx += blockDim.x) {
    int s = idx & 3;        // state
    int y = idx >> 2;       // vocab id
#if HMM_ASYNC_LDS
    __builtin_amdgcn_global_load_async_to_lds_b32(
        (hmm_gptr)(log_B + s * HMM_V + y),
        (hmm_lptr)&ldsB[idx],
        0, 0);
#else
    ldsB[idx] = log_B[s * HMM_V + y];
#endif
  }
  hmm_wait_async_lds();
  __syncthreads();

  const int tid = blockIdx.x * blockDim.x + threadIdx.x;
  const int j = tid & 3;          // state handled by this lane
  const int n = tid >> 2;         // sequence handled by this quad

  // Pre-permuted transition weights (row = previous state, col = j).
  const float Ws  = log_A[j * HMM_S + j];
  const float Wx1 = log_A[(j ^ 1) * HMM_S + j];
  const float Wx2 = log_A[(j ^ 2) * HMM_S + j];
  const float Wx3 = log_A[(j ^ 3) * HMM_S + j];
  const float pij = log_pi[j];

  const int4*   Y4 = (const int4*)(Y + (size_t)n * HMM_T);
  const float4* M4 = (const float4*)(mask + (size_t)n * HMM_T);
  float4* orow = (float4*)(out + (size_t)n * (HMM_S * HMM_T) + (size_t)j * HMM_T);

  float a;
  {
    int4 yv = Y4[0];
    float4 mv = M4[0];
    float4 ob;
    a = pij + ldsB[yv.x * 4 + j];                                    ob.x = a;
    a = hmm_step(a, Ws, Wx1, Wx2, Wx3, ldsB[yv.y * 4 + j], mv.y);    ob.y = a;
    a = hmm_step(a, Ws, Wx1, Wx2, Wx3, ldsB[yv.z * 4 + j], mv.z);    ob.z = a;
    a = hmm_step(a, Ws, Wx1, Wx2, Wx3, ldsB[yv.w * 4 + j], mv.w);    ob.w = a;
    orow[0] = ob;
  }
#pragma unroll 2
  for (int t4 = 1; t4 < HMM_T / 4; ++t4) {
    int4 yv = Y4[t4];
    float4 mv = M4[t4];
    float4 ob;
    a = hmm_step(a, Ws, Wx1, Wx2, Wx3, ldsB[yv.x * 4 + j], mv.x);    ob.x = a;
    a = hmm_step(a, Ws, Wx1, Wx2, Wx3, ldsB[yv.y * 4 + j], mv.y);    ob.y = a;
    a = hmm_step(a, Ws, Wx1, Wx2, Wx3, ldsB[yv.z * 4 + j], mv.z);    ob.z = a;
    a = hmm_step(a, Ws, Wx1, Wx2, Wx3, ldsB[yv.w * 4 + j], mv.w);    ob.w = a;
    orow[t4] = ob;
  }
}

extern "C" void kernel_launch(void* const* d_in, const int* in_sizes, int n_in,
                              void* d_out, int out_size, void* d_ws, size_t ws_size,
                              hipStream_t stream) {
  (void)n_in; (void)out_size; (void)d_ws; (void)ws_size;
  const int*   Y      = (const int*)d_in[0];
  const float* mask   = (const float*)d_in[1];
  const float* log_pi = (const float*)d_in[2];
  const float* log_A  = (const float*)d_in[3];
  const float* log_B  = (const float*)d_in[4];
  float* out = (float*)d_out;

  const int N = in_sizes[0] / HMM_T;      // 16384
  const int threads = N * HMM_S;          // one lane per (sequence, state)
  dim3 block(256);
  dim3 grid((threads + 255) / 256);
  hmm_fwd_kernel<<<grid, block, 0, stream>>>(Y, mask, log_pi, log_A, log_B, out);
}